// CapsuleLayer_75797582840413
// MI455X (gfx1250) — compile-verified
//
#include <hip/hip_runtime.h>

typedef float v2f __attribute__((ext_vector_type(2)));
typedef float v8f __attribute__((ext_vector_type(8)));

#define NB   32     // batch
#define NI   2048   // input capsules
#define DIN  16
#define NJ   32     // output capsules
#define DOUT 32
#define EPS  1e-7f
#define RSTRIDE 20  // padded LDS row stride (dwords): halves hit disjoint banks

// Compute inputs_hat tile for one (j,i): D = Xtile(32x16) * Wtile(16x32) via
// V_WMMA_F32_16X16X4_F32.  acc[t], t = 2*m + n : m = b-tile (16 rows), n = d-tile.
// A-layout (ISA 7.12.2, 32-bit A 16x4): lane<16 -> (M=lane, K=kb..kb+1) = float2.
__device__ __forceinline__ void ihat_tiles(const float* __restrict__ x,
                                           const float* __restrict__ W,
                                           int j, int i, int lane, v8f acc[4]) {
  const int half = lane >> 4;   // 0/1 -> K offset 0/2
  const int l15  = lane & 15;
#pragma unroll
  for (int ks = 0; ks < 4; ++ks) {
    const int kb = ks * 4 + half * 2;
    // A fragments from x[b, i, k]  (k, k+1 contiguous -> one 8B load each)
    const size_t xoff = (size_t)i * DIN + kb;
    v2f a0 = *(const v2f*)(x + (size_t)(l15 +  0) * (NI * DIN) + xoff);
    v2f a1 = *(const v2f*)(x + (size_t)(l15 + 16) * (NI * DIN) + xoff);
    // B fragments from W[j, i, k, d]  (rows k, k+1; cols l15 / l15+16)
    const float* wp = W + ((size_t)(j * NI + i) * DIN + kb) * DOUT + l15;
    v2f b0, b1;
    b0.x = wp[0];   b0.y = wp[DOUT];
    b1.x = wp[16];  b1.y = wp[16 + DOUT];
    acc[0] = __builtin_amdgcn_wmma_f32_16x16x4_f32(false, a0, false, b0, (short)0, acc[0], false, false);
    acc[1] = __builtin_amdgcn_wmma_f32_16x16x4_f32(false, a0, false, b1, (short)0, acc[1], false, false);
    acc[2] = __builtin_amdgcn_wmma_f32_16x16x4_f32(false, a1, false, b0, (short)0, acc[2], false, false);
    acc[3] = __builtin_amdgcn_wmma_f32_16x16x4_f32(false, a1, false, b1, (short)0, acc[3], false, false);
  }
}

// Pass L: logits[b,j,i] = sum_d ihat[b,j,i,d] * O[b,j,d]
// Row-sum via LDS transpose: lane writes partials to its row and column slot in LDS, then
// reads back its own row and sums -> one fully-active coalesced store per i.
__global__ void __launch_bounds__(128) capsL_logits(const float* __restrict__ x,
                                                    const float* __restrict__ W,
                                                    const float* __restrict__ Oacc,
                                                    float* __restrict__ logits) {
  __shared__ float red[4][NB * RSTRIDE];   // per-wave scratch, 10.2 KB total
  const int lane = threadIdx.x & 31;
  const int wave = threadIdx.x >> 5;
  const int j     = blockIdx.x;
  const int ibase = blockIdx.y * 32;
  float* myred = &red[wave][0];

  // Preload O fragments matching the C-tile layout (independent of i -> hoisted)
  float of[4][8];
#pragma unroll
  for (int t = 0; t < 4; ++t) {
    const int bb = (t >> 1) * 16 + (lane >> 4) * 8;
    const int d  = (t & 1) * 16 + (lane & 15);
#pragma unroll
    for (int v = 0; v < 8; ++v)
      of[t][v] = Oacc[(size_t)((bb + v) * NJ + j) * DOUT + d];
  }

  for (int it = 0; it < 8; ++it) {
    const int i = ibase + it * 4 + wave;
    v8f z = {0.f,0.f,0.f,0.f,0.f,0.f,0.f,0.f};
    v8f acc[4] = {z, z, z, z};
    ihat_tiles(x, W, j, i, lane, acc);
    // Scatter partial dot products: p covers this lane's 2 d-values of row R
#pragma unroll
    for (int m = 0; m < 2; ++m) {
#pragma unroll
      for (int v = 0; v < 8; ++v) {
        const float p = acc[2*m][v] * of[2*m][v] + acc[2*m+1][v] * of[2*m+1][v];
        const int R = m * 16 + (lane >> 4) * 8 + v;
        myred[R * RSTRIDE + (lane & 15)] = p;
      }
    }
    // Gather: lane L sums row L (same-wave LDS is in-order, no barrier needed)
    float sum = 0.f;
    const float* rowp = myred + lane * RSTRIDE;
#pragma unroll
    for (int c = 0; c < 16; c += 4) {
      const float4 q = *(const float4*)(rowp + c);
      sum += (q.x + q.y) + (q.z + q.w);
    }
    logits[(size_t)lane * (NJ * NI) + (size_t)j * NI + i] = sum;
  }
}

// Pass S: s[b,j,d] += sum_i c[b,j,i] * ihat[b,j,i,d]
__global__ void __launch_bounds__(128) capsS_outputs(const float* __restrict__ x,
                                                     const float* __restrict__ W,
                                                     const float* __restrict__ att,
                                                     float* __restrict__ s_buf,
                                                     int uniform_c) {
  const int lane = threadIdx.x & 31;
  const int wave = threadIdx.x >> 5;
  const int j     = blockIdx.x;
  const int ibase = blockIdx.y * 32;

  v8f z = {0.f,0.f,0.f,0.f,0.f,0.f,0.f,0.f};
  v8f sacc[4] = {z, z, z, z};

  for (int it = 0; it < 8; ++it) {
    const int i = ibase + it * 4 + wave;
    v8f acc[4] = {z, z, z, z};
    ihat_tiles(x, W, j, i, lane, acc);
    float cf[2][8];
    if (uniform_c) {
#pragma unroll
      for (int m = 0; m < 2; ++m)
#pragma unroll
        for (int v = 0; v < 8; ++v) cf[m][v] = 1.0f / NJ;
    } else {
#pragma unroll
      for (int m = 0; m < 2; ++m) {
#pragma unroll
        for (int v = 0; v < 8; ++v) {
          const int b = m * 16 + (lane >> 4) * 8 + v;
          cf[m][v] = att[(size_t)b * (NJ * NI) + (size_t)j * NI + i]; // broadcast
        }
      }
    }
#pragma unroll
    for (int t = 0; t < 4; ++t) {
      const int m = t >> 1;
#pragma unroll
      for (int v = 0; v < 8; ++v)
        sacc[t][v] += cf[m][v] * acc[t][v];
    }
  }
  // fold per-wave partials into s_buf (once per wave, 32 atomics)
#pragma unroll
  for (int t = 0; t < 4; ++t) {
    const int bb = (t >> 1) * 16 + (lane >> 4) * 8;
    const int d  = (t & 1) * 16 + (lane & 15);
#pragma unroll
    for (int v = 0; v < 8; ++v)
      atomicAdd(&s_buf[(size_t)((bb + v) * NJ + j) * DOUT + d], sacc[t][v]);
  }
}

// In-place softmax over j for each (b,i) column (stride NI)
__global__ void caps_softmax(float* __restrict__ cbuf) {
  const int idx = blockIdx.x * blockDim.x + threadIdx.x;
  if (idx >= NB * NI) return;
  const int b = idx >> 11;          // / NI
  const int i = idx & (NI - 1);
  float* p = cbuf + (size_t)b * (NJ * NI) + i;
  float e[NJ];
  float mx = -3.402823466e38f;
#pragma unroll
  for (int jj = 0; jj < NJ; ++jj) { e[jj] = p[(size_t)jj * NI]; mx = fmaxf(mx, e[jj]); }
  float s = 0.f;
#pragma unroll
  for (int jj = 0; jj < NJ; ++jj) { e[jj] = __expf(e[jj] - mx); s += e[jj]; }
  const float inv = 1.f / s;
#pragma unroll
  for (int jj = 0; jj < NJ; ++jj) p[(size_t)jj * NI] = e[jj] * inv;
}

// Squash + maintain running O = sum of previous outputs; final round writes d_out
__global__ void caps_squash(const float* __restrict__ s, float* __restrict__ Oacc,
                            float* __restrict__ outbuf, int r) {
  const int idx = blockIdx.x * blockDim.x + threadIdx.x;   // b*NJ + j
  if (idx >= NB * NJ) return;
  float v[DOUT];
  float sq = 0.f;
#pragma unroll
  for (int d = 0; d < DOUT; ++d) { v[d] = s[(size_t)idx * DOUT + d]; sq += v[d] * v[d]; }
  const float scale = sq / (1.f + sq) * rsqrtf(sq + EPS);
#pragma unroll
  for (int d = 0; d < DOUT; ++d) {
    const float o = scale * v[d];
    if (r == 2)      outbuf[(size_t)idx * DOUT + d] = o;
    else if (r == 0) Oacc[(size_t)idx * DOUT + d] = o;
    else             Oacc[(size_t)idx * DOUT + d] += o;
  }
}

__global__ void caps_zero(float* __restrict__ p, int n) {
  const int idx = blockIdx.x * blockDim.x + threadIdx.x;
  if (idx < n) p[idx] = 0.f;
}

extern "C" void kernel_launch(void* const* d_in, const int* in_sizes, int n_in,
                              void* d_out, int out_size, void* d_ws, size_t ws_size,
                              hipStream_t stream) {
  (void)in_sizes; (void)n_in; (void)out_size; (void)ws_size;
  const float* x = (const float*)d_in[0];   // [B, I, DIN]
  const float* W = (const float*)d_in[1];   // [J, I, DIN, DOUT]
  float* out = (float*)d_out;               // [B, J, DOUT]

  float* cbuf  = (float*)d_ws;                          // [B, J, I]  (logits -> c)
  float* s_buf = cbuf  + (size_t)NB * NJ * NI;          // [B, J, DOUT]
  float* Oacc  = s_buf + (size_t)NB * NJ * DOUT;        // [B, J, DOUT]

  const dim3 gemm_grid(NJ, NI / 32);                    // 32 x 64 blocks, 4 waves each
  const int SN = NB * NJ * DOUT;                        // 32768

  for (int r = 0; r < 3; ++r) {
    if (r > 0) {
      capsL_logits<<<gemm_grid, 128, 0, stream>>>(x, W, Oacc, cbuf);
      caps_softmax<<<(NB * NI + 255) / 256, 256, 0, stream>>>(cbuf);
    }
    caps_zero<<<(SN + 255) / 256, 256, 0, stream>>>(s_buf, SN);
    capsS_outputs<<<gemm_grid, 128, 0, stream>>>(x, W, cbuf, s_buf, (r == 0) ? 1 : 0);
    caps_squash<<<(NB * NJ + 255) / 256, 256, 0, stream>>>(s_buf, Oacc, out, r);
  }
}